// JointAttention_2190433321651
// MI455X (gfx1250) — compile-verified
//
#include <hip/hip_runtime.h>
#include <math.h>

// Problem constants
#define BB 64
#define NN 8192
#define DD 64
#define JJ 13
#define SPLITS 8
#define NS (NN / SPLITS)      // 1024 tokens per split
#define TILE 64               // tokens per chunk
#define NCHUNK (NS / TILE)    // 16

typedef __attribute__((ext_vector_type(2))) float v2f;
typedef __attribute__((ext_vector_type(8))) float v8f;

// ---- dynamic LDS layout (floats), total ~95 KB (<320 KB/WGP) ----
#define XS_PITCH 68
#define KV_PITCH 132
#define PS_PITCH 68
#define XS0_OFF  0
#define XSBUF    (64*XS_PITCH)                   // 4352 (triple buffered)
#define KV_OFF   (3*XSBUF)                       // 13056
#define LS_OFF   (KV_OFF + 64*KV_PITCH)          // 21504
#define PS_OFF   (LS_OFF + 64*16)                // 22528  (P stored [j][n])
#define MR_OFF   (PS_OFF + 16*PS_PITCH)          // 23616
#define SR_OFF   (MR_OFF + 16)
#define SC_OFF   (SR_OFF + 16)
#define SMEM_FLOATS (SC_OFF + 16)                // 23664 floats = 94656 B

// ACC2: scratch for cross-wave accumulator reduction (reuses KV area after loop)
#define ACC2_OFF KV_OFF

__device__ __forceinline__ v8f wmma4(v2f a, v2f b, v8f c) {
  // D = A(16x4,f32) * B(4x16,f32) + C(16x16,f32)
  return __builtin_amdgcn_wmma_f32_16x16x4_f32(
      /*neg_a=*/false, a, /*neg_b=*/false, b,
      /*c_mod=*/(short)0, c, /*reuse_a=*/false, /*reuse_b=*/false);
}

// Async global -> LDS copy of one 16B packet (GVS mode: saddr + 32-bit voffset).
// Writes LDS directly, tracked by ASYNCcnt — no VGPR data round-trip.
__device__ __forceinline__ void async_b128(unsigned lds_addr, unsigned goff,
                                           const float* sbase) {
  asm volatile("global_load_async_to_lds_b128 %0, %1, %2"
               :: "v"(lds_addr), "v"(goff), "s"(sbase) : "memory");
}
// Wait until at most N async ops outstanding (ASYNCcnt decrements in order).
__device__ __forceinline__ void wait_async_le4() {
  asm volatile("s_wait_asynccnt 4" ::: "memory");
}

__global__ __launch_bounds__(256)
void ja_main(const float* __restrict__ x, const float* __restrict__ emb,
             const float* __restrict__ Wkv, const float* __restrict__ bkv,
             float* __restrict__ pm, float* __restrict__ psum,
             float* __restrict__ pacc)
{
  extern __shared__ float sm[];
  const int tid  = threadIdx.x;
  const int lane = tid & 31;
  const int wave = tid >> 5;          // 8 waves (wave32)
  const int l16  = lane & 15;
  const int hi   = lane >> 4;         // half-wave select
  const int kk   = hi << 1;           // K sub-offset {0,2}

  const int b  = blockIdx.x / SPLITS;
  const int sp = blockIdx.x % SPLITS;
  const float* xg = x + ((long)b * NN + (long)sp * NS) * DD;

  // LDS byte base of dynamic shared (flat LDS address low 32 bits == offset)
  const unsigned ldsb = (unsigned)(unsigned long long)(uintptr_t)sm;

  // ---- async-stage x chunks 0 and 1 (prefetch depth 2, 4 packets/wave each) ----
  #pragma unroll
  for (int pc = 0; pc < 2; ++pc) {
    const unsigned gbase = (unsigned)(pc * TILE * DD) * 4u;
    const int xo = XS0_OFF + pc * XSBUF;
    for (int i = tid; i < (TILE*DD)/4; i += 256) {
      int r = i >> 4, c4 = i & 15;
      async_b128(ldsb + (unsigned)(xo + r*XS_PITCH + c4*4)*4u,
                 gbase + (unsigned)(r*DD + c4*4)*4u, xg);
    }
  }

  if (tid < 16) { sm[MR_OFF+tid] = -3.0e38f; sm[SR_OFF+tid] = 0.f; sm[SC_OFF+tid] = 0.f; }

  // GEMM1 ownership: wave w -> 2x2 block of 16x16 tiles
  //   row blocks rb0, rb0+1 ; col blocks cb, cb+1
  const int rb0 = (wave & 1) * 2;
  const int cb  = (wave >> 1) * 2;
  float bias[2];
  #pragma unroll
  for (int c = 0; c < 2; ++c) bias[c] = bkv[(cb + c)*16 + l16];

  // ---- hoist loop-invariant B operands into registers ----
  // W_kv in WMMA-B layout: wreg[c][k2] = rows (4*k2+kk, +1), col (cb+c)*16+l16
  v2f wreg[2][16];                     // 64 VGPRs
  #pragma unroll
  for (int c = 0; c < 2; ++c) {
    #pragma unroll
    for (int k2 = 0; k2 < 16; ++k2) {
      const int k0 = k2 * 4;
      wreg[c][k2].x = Wkv[(k0+kk  )*128 + (cb+c)*16 + l16];
      wreg[c][k2].y = Wkv[(k0+kk+1)*128 + (cb+c)*16 + l16];
    }
  }
  // E^T in WMMA-B layout: ereg[k2] = E^T rows (4*k2+kk, +1), col j=l16 (0 if j>=J)
  v2f ereg[16];                        // 32 VGPRs
  {
    const bool jok = (l16 < JJ);
    #pragma unroll
    for (int k2 = 0; k2 < 16; ++k2) {
      const int k0 = k2 * 4;
      ereg[k2].x = jok ? emb[l16*DD + k0 + kk    ] : 0.f;
      ereg[k2].y = jok ? emb[l16*DD + k0 + kk + 1] : 0.f;
    }
  }

  // GEMM3 ownership: all 8 waves; pairs (w, w+4) share a d-tile, split K range
  v8f acc = {};                       // persistent out tile (j x [d0,d0+16))
  const int d0 = (wave & 3) * 16;
  const int g3k = (wave >> 2) * 32;   // K half: [g3k, g3k+32)

  for (int ch = 0; ch < NCHUNK; ++ch) {
    const int xo = XS0_OFF + (ch % 3) * XSBUF;

    wait_async_le4();                 // chunk ch resident (ch+1 may still fly)
    __syncthreads();

    // ---- GEMM1: kv = x @ W_kv + b  -> KVs[64 x 128] (2x2 tiles per wave) ----
    {
      v8f cc[2][2];
      #pragma unroll
      for (int ri = 0; ri < 2; ++ri)
        #pragma unroll
        for (int c = 0; c < 2; ++c)
          #pragma unroll
          for (int r = 0; r < 8; ++r) cc[ri][c][r] = bias[c];

      #pragma unroll
      for (int k2 = 0; k2 < 16; ++k2) {
        const int k0 = k2 * 4;
        v2f a0 = *(const v2f*)&sm[xo + ((rb0  )*16 + l16)*XS_PITCH + k0 + kk];
        v2f a1 = *(const v2f*)&sm[xo + ((rb0+1)*16 + l16)*XS_PITCH + k0 + kk];
        #pragma unroll
        for (int c = 0; c < 2; ++c) {
          cc[0][c] = wmma4(a0, wreg[c][k2], cc[0][c]);
          cc[1][c] = wmma4(a1, wreg[c][k2], cc[1][c]);
        }
      }
      #pragma unroll
      for (int ri = 0; ri < 2; ++ri)
        #pragma unroll
        for (int c = 0; c < 2; ++c)
          #pragma unroll
          for (int r = 0; r < 8; ++r)
            sm[KV_OFF + ((rb0+ri)*16 + r + 8*hi)*KV_PITCH + (cb+c)*16 + l16] =
                cc[ri][c][r];
    }
    __syncthreads();

    // ---- issue async load of chunk ch+2 into buffer (ch+2)%3 ----
    // (overlaps with GEMM2 + softmax + GEMM3 below AND next chunk's wait)
    if (ch + 2 < NCHUNK) {
      const int nxo = XS0_OFF + ((ch + 2) % 3) * XSBUF;
      const unsigned gbase = (unsigned)((ch + 2) * TILE * DD) * 4u;
      for (int i = tid; i < (TILE*DD)/4; i += 256) {
        int r = i >> 4, c4 = i & 15;
        async_b128(ldsb + (unsigned)(nxo + r*XS_PITCH + c4*4)*4u,
                   gbase + (unsigned)(r*DD + c4*4)*4u, xg);
      }
    }

    // ---- GEMM2: logits = k @ E^T  -> Ls[64 x 16]  (waves 0..3) ----
    if (wave < 4) {
      v8f lg = {};
      #pragma unroll
      for (int k2 = 0; k2 < 16; ++k2) {
        const int k0 = k2 * 4;
        v2f a = *(const v2f*)&sm[KV_OFF + (wave*16 + l16)*KV_PITCH + k0 + kk];
        lg = wmma4(a, ereg[k2], lg);
      }
      #pragma unroll
      for (int r = 0; r < 8; ++r)
        sm[LS_OFF + (wave*16 + r + 8*hi)*16 + l16] = lg[r];
    }
    __syncthreads();

    // ---- online column-softmax update (one thread per j) ----
    if (tid < 16) {
      float mold = sm[MR_OFF + tid];
      float tmax = -3.0e38f;
      for (int n = 0; n < TILE; ++n) tmax = fmaxf(tmax, sm[LS_OFF + n*16 + tid]);
      float mnew = fmaxf(mold, tmax);
      float scl  = __expf(mold - mnew);     // 0 on first chunk
      float ssum = 0.f;
      for (int n = 0; n < TILE; ++n) {
        float e = __expf(sm[LS_OFF + n*16 + tid] - mnew);
        sm[PS_OFF + tid*PS_PITCH + n] = e;  // P stored transposed: [j][n]
        ssum += e;
      }
      sm[MR_OFF + tid] = mnew;
      sm[SR_OFF + tid] = sm[SR_OFF + tid] * scl + ssum;
      sm[SC_OFF + tid] = scl;
    }
    __syncthreads();

    // ---- rescale + GEMM3: acc += P^T @ v  (all 8 waves, K split in halves) ----
    {
      #pragma unroll
      for (int r = 0; r < 8; ++r) acc[r] *= sm[SC_OFF + r + 8*hi];
      #pragma unroll
      for (int k2 = 0; k2 < 8; ++k2) {
        const int k0 = g3k + k2 * 4;
        // A[M=j=l16][K=token]: contiguous pair in transposed P
        v2f a = *(const v2f*)&sm[PS_OFF + l16*PS_PITCH + k0 + kk];
        v2f bv;                                 // B[K=token][N=d]
        bv.x = sm[KV_OFF + (k0+kk  )*KV_PITCH + 64 + d0 + l16];
        bv.y = sm[KV_OFF + (k0+kk+1)*KV_PITCH + 64 + d0 + l16];
        acc = wmma4(a, bv, acc);
      }
    }
    // loop-top wait+barrier also separates GEMM3's KV reads from next GEMM1 writes
  }
  __syncthreads();

  // ---- reduce the two K-half accumulators (waves 4..7 -> waves 0..3) ----
  if (wave >= 4) {
    #pragma unroll
    for (int r = 0; r < 8; ++r)
      sm[ACC2_OFF + (r + 8*hi)*64 + d0 + l16] = acc[r];
  }
  __syncthreads();

  // ---- write split partials ----
  const long pb = (long)(b * SPLITS + sp);
  if (tid < 16) {
    pm  [pb*16 + tid] = sm[MR_OFF + tid];
    psum[pb*16 + tid] = sm[SR_OFF + tid];
  }
  if (wave < 4) {
    #pragma unroll
    for (int r = 0; r < 8; ++r) {
      float v = acc[r] + sm[ACC2_OFF + (r + 8*hi)*64 + d0 + l16];
      pacc[(pb*16 + r + 8*hi)*64 + d0 + l16] = v;
    }
  }
}

// combine S split partials: standard log-sum-exp merge + normalize
__global__ __launch_bounds__(64)
void ja_combine(const float* __restrict__ pm, const float* __restrict__ psum,
                const float* __restrict__ pacc, float* __restrict__ out)
{
  const int bj = blockIdx.x;          // B*J
  const int b = bj / JJ, j = bj % JJ;
  const int d = threadIdx.x;          // 64
  float m = -3.0e38f;
  for (int s = 0; s < SPLITS; ++s)
    m = fmaxf(m, pm[(b*SPLITS + s)*16 + j]);
  float tot = 0.f, val = 0.f;
  for (int s = 0; s < SPLITS; ++s) {
    float w = __expf(pm[(b*SPLITS + s)*16 + j] - m);
    tot += w * psum[(b*SPLITS + s)*16 + j];
    val += w * pacc[((b*SPLITS + s)*16 + j)*64 + d];
  }
  out[(b*JJ + j)*DD + d] = val / tot;
}

extern "C" void kernel_launch(void* const* d_in, const int* in_sizes, int n_in,
                              void* d_out, int out_size, void* d_ws, size_t ws_size,
                              hipStream_t stream) {
  (void)in_sizes; (void)n_in; (void)out_size; (void)ws_size;
  const float* x   = (const float*)d_in[0];
  const float* emb = (const float*)d_in[1];
  const float* Wkv = (const float*)d_in[2];
  const float* bkv = (const float*)d_in[3];
  float* out = (float*)d_out;

  float* pm   = (float*)d_ws;                    // B*S*16
  float* psum = pm   + BB*SPLITS*16;             // B*S*16
  float* pacc = psum + BB*SPLITS*16;             // B*S*16*64

  const size_t smem = SMEM_FLOATS * sizeof(float);   // ~95 KB
  hipFuncSetAttribute(reinterpret_cast<const void*>(ja_main),
                      hipFuncAttributeMaxDynamicSharedMemorySize, (int)smem);

  ja_main<<<BB*SPLITS, 256, smem, stream>>>(x, emb, Wkv, bkv, pm, psum, pacc);
  ja_combine<<<BB*JJ, 64, 0, stream>>>(pm, psum, pacc, out);
}